// SplineConv_16149077033177
// MI455X (gfx1250) — compile-verified
//
#include <hip/hip_runtime.h>

// SplineConv for MI455X (gfx1250, wave32).
// Phase 1: wave-per-edge scatter of basis-weighted source features into a
//          [N*KP, 64] f32 accumulator with hardware global_atomic_add_f32.
// Phase 2: 25-way batched GEMM acc[n,k,:] @ W[k] with V_WMMA_F32_16X16X4_F32.
//          W[k] is staged into a double-buffered LDS tile by the Tensor Data
//          Mover (TENSOR_LOAD_TO_LDS, TENSORcnt) overlapped with WMMA compute;
//          falls back to cooperative vector copy when the TDM builtin is
//          unavailable. Root-weight GEMM is folded in as a 26th pipeline stage.

#define KDIM 5
#define KP 25
#define IN_F 64
#define OUT_F 64

typedef __attribute__((ext_vector_type(2))) float v2f;
typedef __attribute__((ext_vector_type(8))) float v8f;
typedef __attribute__((ext_vector_type(4))) unsigned int u32x4;
typedef __attribute__((ext_vector_type(4))) int i32x4;
typedef __attribute__((ext_vector_type(8))) int i32x8;

#if __has_builtin(__builtin_amdgcn_tensor_load_to_lds) && \
    __has_builtin(__builtin_amdgcn_s_wait_tensorcnt)
#define SPLINE_HAS_TDM 1
#else
#define SPLINE_HAS_TDM 0
#endif

// ---------------------------------------------------------------------------
// Stage one 16 KB tile (64x64 f32) from global into LDS.
// TDM path: wave 0 issues a single descriptor-driven DMA (async, TENSORcnt).
// Fallback: cooperative float4 copy by all 256 threads.
// ---------------------------------------------------------------------------
__device__ __forceinline__ void stage_issue(float* __restrict__ dst,
                                            const float* __restrict__ src,
                                            int tid) {
#if SPLINE_HAS_TDM
  if ((tid >> 5) == 0) {  // one wave issues; TDM ignores EXEC
    // LDS aperture: generic address low 32 bits == LDS byte offset.
    const unsigned int lds_off = (unsigned int)(uintptr_t)dst;
    const unsigned long long ga = (unsigned long long)(uintptr_t)src;
    u32x4 g0;
    g0[0] = 1u;                                   // count=1 (valid user D#)
    g0[1] = lds_off;                              // lds_addr
    g0[2] = (unsigned int)ga;                     // global_addr[31:0]
    g0[3] = (unsigned int)(ga >> 32) | (2u << 30);// global_addr[56:32]|type=2
    i32x8 g1;
    g1[0] = (int)(2u << 16);       // workgroup_mask=0, data_size=2 (4 bytes)
    g1[1] = (int)(4096u << 16);    // tensor_dim0[15:0]=4096 (in [31:16])
    g1[2] = (int)(1u << 16);       // tensor_dim0[31:16]=0, tensor_dim1 lo=1
    g1[3] = (int)(4096u << 16);    // tensor_dim1 hi=0, tile_dim0=4096
    g1[4] = 0;                     // tile_dim1=0 (unused), tile_dim2=0
    g1[5] = 4096;                  // tensor_dim0_stride lo32
    g1[6] = 0;
    g1[7] = 0;
    i32x4 z4 = {0, 0, 0, 0};
#if __clang_major__ >= 23
    i32x8 z8 = {0, 0, 0, 0, 0, 0, 0, 0};
    __builtin_amdgcn_tensor_load_to_lds(g0, g1, z4, z4, z8, 0);
#else
    __builtin_amdgcn_tensor_load_to_lds(g0, g1, z4, z4, 0);
#endif
  }
#else
  const float4* s4 = (const float4*)src;
  float4* d4 = (float4*)dst;
#pragma unroll
  for (int r = 0; r < 4; ++r) d4[tid + 256 * r] = s4[tid + 256 * r];
  __builtin_prefetch(src + IN_F * OUT_F, 0, 1);
#endif
}

__device__ __forceinline__ void stage_wait(int tid) {
#if SPLINE_HAS_TDM
  if ((tid >> 5) == 0) __builtin_amdgcn_s_wait_tensorcnt(0);
#endif
  __syncthreads();  // publish staged LDS tile to all waves
}

// ---------------------------------------------------------------------------
// Phase 1: edge scatter. One wave (32 lanes) per edge; each lane owns feature
// pair (lane, lane+32). 8 edges per 256-thread block.
// ---------------------------------------------------------------------------
__global__ void spline_scatter_kernel(const float* __restrict__ x,
                                      const int* __restrict__ edge_index,
                                      const float* __restrict__ pseudo,
                                      float* __restrict__ acc,
                                      float* __restrict__ deg,
                                      int E) {
  const int wave = threadIdx.x >> 5;
  const int lane = threadIdx.x & 31;
  const long e = (long)blockIdx.x * 8 + wave;
  if (e >= E) return;

  const int row = edge_index[e];            // target
  const int col = edge_index[(long)E + e];  // source

  // degree-1 open B-spline: v = pseudo * (K-1)
  const float v0 = pseudo[2 * e + 0] * (float)(KDIM - 1);
  const float v1 = pseudo[2 * e + 1] * (float)(KDIM - 1);
  const float fl0 = floorf(v0), fl1 = floorf(v1);
  const float f0 = v0 - fl0, f1 = v1 - fl1;
  const int l0 = (int)fl0, l1 = (int)fl1;

  const float* xr = x + (long)col * IN_F;
  const float xa = xr[lane];
  const float xb = xr[lane + 32];

  float* accrow = acc + (long)row * KP * IN_F;

#pragma unroll
  for (int s = 0; s < 4; ++s) {
    const int b0 = s & 1;
    const int b1 = (s >> 1) & 1;
    const float w = (b0 ? f0 : 1.0f - f0) * (b1 ? f1 : 1.0f - f1);
    const int i0 = (l0 + b0) % KDIM;
    const int i1 = (l1 + b1) % KDIM;
    const int idx = i0 * KDIM + i1;  // row-major flatten: offs = {5,1}
    float* dst = accrow + idx * IN_F;
    unsafeAtomicAdd(dst + lane, w * xa);
    unsafeAtomicAdd(dst + lane + 32, w * xb);
  }
  if (lane == 0) unsafeAtomicAdd(deg + row, 1.0f);
}

// ---------------------------------------------------------------------------
// Phase 2: out[n,:] = (1/deg) * sum_k acc[n,k,:] @ W[k]  + x[n,:] @ R + bias
// 8 waves per block; wave w owns node tile [tile*16, tile*16+16).
// Double-buffered LDS: stage W[k+1] (or root for k=24) while computing on W[k].
//
// A layout (16x4 f32): lanes 0-15 hold M=0..15 with K={0,1}, lanes 16-31 hold
// M=0..15 with K={2,3} -> per-lane contiguous float2 from acc row.
// B layout (4x16 f32): VGPR0 = rows K0/K2 per lane-half, VGPR1 = K1/K3,
// N = lane&15.  C/D layout: VGPR r holds M = r + 8*(lane>>4), N = lane&15.
// ---------------------------------------------------------------------------
__global__ void spline_gemm_kernel(const float* __restrict__ acc,
                                   const float* __restrict__ deg,
                                   const float* __restrict__ x,
                                   const float* __restrict__ weight,
                                   const float* __restrict__ rootw,
                                   const float* __restrict__ bias,
                                   float* __restrict__ out,
                                   int ntiles) {
  __shared__ float lds_w[2][IN_F * OUT_F];  // 2 x 16 KB double buffer

  const int tid = threadIdx.x;
  const int lane = tid & 31;
  const int wave = tid >> 5;
  const int tile = blockIdx.x * 8 + wave;
  const bool valid = (tile < ntiles);  // wave-uniform
  const int nbase = tile * 16;

  const int mlane = lane & 15;  // M for A, N for B/C/D
  const int khalf = lane >> 4;  // selects K pair {0,1} vs {2,3}

  v8f c[4];
#pragma unroll
  for (int t = 0; t < 4; ++t)
#pragma unroll
    for (int r = 0; r < 8; ++r) c[t][r] = 0.0f;

  // Prologue: stage W[0] into buffer 0.
  stage_issue(&lds_w[0][0], weight, tid);
  stage_wait(tid);

  for (int k = 0; k < KP; ++k) {
    // Stage next tile (W[k+1], or root_weight as virtual stage k=25) into the
    // buffer not being read this iteration; overlaps with WMMA below.
    const float* nsrc =
        (k + 1 < KP) ? (weight + (long)(k + 1) * IN_F * OUT_F) : rootw;
    stage_issue(&lds_w[(k + 1) & 1][0], nsrc, tid);

    if (valid) {
      const float* __restrict__ wb = &lds_w[k & 1][0];
      const float* arow = acc + ((long)(nbase + mlane) * KP + k) * IN_F;
#pragma unroll
      for (int i0 = 0; i0 < IN_F; i0 += 4) {
        const int ka = i0 + 2 * khalf;
        v2f a;
        a.x = arow[ka + 0];
        a.y = arow[ka + 1];
#pragma unroll
        for (int t = 0; t < 4; ++t) {
          const int o = t * 16 + mlane;
          v2f b;
          b.x = wb[(ka + 0) * OUT_F + o];
          b.y = wb[(ka + 1) * OUT_F + o];
          c[t] = __builtin_amdgcn_wmma_f32_16x16x4_f32(
              false, a, false, b, (short)0, c[t], false, false);
        }
      }
    }
    stage_wait(tid);  // TDM done (wave 0) + block barrier
  }

  // root_weight now resides in lds_w[(KP)&1] == lds_w[1].
  if (valid) {
    const float* __restrict__ wb = &lds_w[1][0];

    // 1/max(deg,1) for the 8 node rows this lane's C registers cover.
    float invd[8];
#pragma unroll
    for (int r = 0; r < 8; ++r) {
      const float dg = deg[nbase + 8 * khalf + r];
      invd[r] = 1.0f / fmaxf(dg, 1.0f);
    }

    const float* xrow = x + (long)(nbase + mlane) * IN_F;
#pragma unroll
    for (int t = 0; t < 4; ++t) {
      const int o = t * 16 + mlane;
      const float bv = bias[o];
      v8f cc;
#pragma unroll
      for (int r = 0; r < 8; ++r) cc[r] = c[t][r] * invd[r] + bv;
      // fused root-weight GEMM: cc += x_tile @ R (column block o-tile)
#pragma unroll
      for (int i0 = 0; i0 < IN_F; i0 += 4) {
        const int ka = i0 + 2 * khalf;
        v2f a;
        a.x = xrow[ka + 0];
        a.y = xrow[ka + 1];
        v2f b;
        b.x = wb[(ka + 0) * OUT_F + o];
        b.y = wb[(ka + 1) * OUT_F + o];
        cc = __builtin_amdgcn_wmma_f32_16x16x4_f32(
            false, a, false, b, (short)0, cc, false, false);
      }
#pragma unroll
      for (int r = 0; r < 8; ++r)
        out[(long)(nbase + 8 * khalf + r) * OUT_F + o] = cc[r];
    }
  }
}

// ---------------------------------------------------------------------------
extern "C" void kernel_launch(void* const* d_in, const int* in_sizes, int n_in,
                              void* d_out, int out_size, void* d_ws, size_t ws_size,
                              hipStream_t stream) {
  const float* x = (const float*)d_in[0];       // [N, 64]
  const int* edge_index = (const int*)d_in[1];  // [2, E]
  const float* pseudo = (const float*)d_in[2];  // [E, 2]
  const float* weight = (const float*)d_in[3];  // [25, 64, 64]
  const float* rootw = (const float*)d_in[4];   // [64, 64]
  const float* bias = (const float*)d_in[5];    // [64]
  float* out = (float*)d_out;                   // [N, 64]

  const int N = in_sizes[0] / IN_F;
  const int E = in_sizes[1] / 2;

  float* acc = (float*)d_ws;                 // [N*KP, 64] f32
  float* deg = acc + (size_t)N * KP * IN_F;  // [N] f32
  const size_t zero_bytes =
      ((size_t)N * KP * IN_F + (size_t)N) * sizeof(float);
  hipMemsetAsync(d_ws, 0, zero_bytes, stream);

  dim3 blk(256);
  const int g1 = (E + 7) / 8;  // 8 edges (waves) per block
  spline_scatter_kernel<<<g1, blk, 0, stream>>>(x, edge_index, pseudo, acc,
                                                deg, E);

  const int ntiles = N / 16;        // N = 50000 -> 3125 tiles
  const int g2 = (ntiles + 7) / 8;  // 8 node-tiles (waves) per block
  spline_gemm_kernel<<<g2, blk, 0, stream>>>(acc, deg, x, weight, rootw, bias,
                                             out, ntiles);
}